// Jitter_8400956031468
// MI455X (gfx1250) — compile-verified
//
#include <hip/hip_runtime.h>
#include <stdint.h>

#define B_ 16
#define I_ 256
#define T_ 8192
#define NDRAW (T_ - 2)   // interior Markov draws per batch
#define R_ 4             // channel rows per gather workgroup (shares srcidx[b])

typedef unsigned int uint32x4_t_ __attribute__((ext_vector_type(4)));
typedef int          int32x8_t_  __attribute__((ext_vector_type(8)));
typedef int          int32x4_t_  __attribute__((ext_vector_type(4)));

__device__ __forceinline__ uint32_t rotl32(uint32_t v, int n) {
  return (v << n) | (v >> (32 - n));
}

// Threefry-2x32, 20 rounds — matches JAX's PRNG core.
__device__ __forceinline__ void threefry2x32(uint32_t k0, uint32_t k1,
                                             uint32_t c0, uint32_t c1,
                                             uint32_t& o0, uint32_t& o1) {
  const uint32_t ks2 = k0 ^ k1 ^ 0x1BD11BDAu;
  uint32_t x0 = c0 + k0, x1 = c1 + k1;
#define QR4(r0, r1, r2, r3)                                                    \
  x0 += x1; x1 = rotl32(x1, r0); x1 ^= x0;                                     \
  x0 += x1; x1 = rotl32(x1, r1); x1 ^= x0;                                     \
  x0 += x1; x1 = rotl32(x1, r2); x1 ^= x0;                                     \
  x0 += x1; x1 = rotl32(x1, r3); x1 ^= x0;
  QR4(13, 15, 26, 6)  x0 += k1;  x1 += ks2 + 1u;
  QR4(17, 29, 16, 24) x0 += ks2; x1 += k0 + 2u;
  QR4(13, 15, 26, 6)  x0 += k0;  x1 += k1 + 3u;
  QR4(17, 29, 16, 24) x0 += k1;  x1 += ks2 + 4u;
  QR4(13, 15, 26, 6)  x0 += ks2; x1 += k0 + 5u;
#undef QR4
  o0 = x0; o1 = x1;
}

// JAX-style bits->uniform(0,1)->Gumbel
__device__ __forceinline__ float bits_to_gumbel(uint32_t bits) {
  uint32_t m = (bits >> 9) | 0x3f800000u;
  float u = __uint_as_float(m) - 1.0f;
  u = fmaxf(u, 1.175494351e-38f);
  return -__logf(-__logf(u));
}

// Kernel A: fully parallel per-(b,step) Gumbel draws + both argmax candidates.
// cand[b*T_+k] = a | (c<<2): a = choice for generic rows, c = choice for (2,1).
__global__ void __launch_bounds__(256)
jitter_gumbel_kernel(const float* __restrict__ probs, uint8_t* __restrict__ cand) {
  int id = blockIdx.x * blockDim.x + threadIdx.x;
  if (id >= B_ * NDRAW) return;
  int b = id / NDRAW;
  int k = id - b * NDRAW;

  // master key (0,42); batch key = split(key, B)[b]
  uint32_t kb0, kb1;
  threefry2x32(0u, 42u, (uint32_t)b, (uint32_t)(b + B_), kb0, kb1);
  // per-step subkey = split(kb, NDRAW)[k]
  uint32_t s0, s1;
  threefry2x32(kb0, kb1, (uint32_t)k, (uint32_t)(k + NDRAW), s0, s1);
  // random_bits(subkey, shape (3,)): counters paired as (0,2) and (1,0);
  // bits = [o0(0,2), o0(1,0), o1(0,2)]
  uint32_t a0, a1, q0, q1;
  threefry2x32(s0, s1, 0u, 2u, a0, a1);
  threefry2x32(s0, s1, 1u, 0u, q0, q1);
  float g0 = bits_to_gumbel(a0);
  float g1 = bits_to_gumbel(q0);
  float g2 = bits_to_gumbel(a1);

  // generic row = probs[0][0][:] = [p, s, p]; special row = probs[2][1][:]
  float lp = __logf(probs[0]);
  float ls = __logf(probs[1]);
  float r1 = probs[2 * 9 + 1 * 3 + 1];
  float r2 = probs[2 * 9 + 1 * 3 + 2];
  float lc1 = __logf(r1), lc2 = __logf(r2);

  float v0 = lp + g0, v1 = ls + g1, v2 = lp + g2;
  int a = 0; float best = v0;
  if (v1 > best) { best = v1; a = 1; }
  if (v2 > best) { a = 2; }

  // row (2,1): category 0 has prob 0 => excluded
  int c = ((lc2 + g2) > (lc1 + g1)) ? 2 : 1;

  cand[b * T_ + k] = (uint8_t)(a | (c << 2));
}

// Kernel B: tiny serial Markov scan, one lane per batch; u32-wide code loads.
// srcidx[b][t] = source column for output position t (in [0, T-1]).
__global__ void jitter_scan_kernel(const uint8_t* __restrict__ cand,
                                   int* __restrict__ srcidx) {
  int b = threadIdx.x;
  if (b >= B_) return;
  const uint32_t* cw = (const uint32_t*)(cand + (size_t)b * T_);  // stride T_ keeps alignment
  int* sb = srcidx + (size_t)b * T_;
  sb[0] = 0;                       // mindex[1] == 1
  int p2 = 1, p1 = 1;
  for (int w = 0; w < T_ / 4; ++w) {
    uint32_t pk4 = cw[w];
#pragma unroll
    for (int e = 0; e < 4; ++e) {
      int k = w * 4 + e;
      if (k < NDRAW) {
        uint32_t pk = (pk4 >> (8 * e)) & 0xffu;
        int j = (p2 == 2 && p1 == 1) ? (int)(pk >> 2) : (int)(pk & 3u);
        sb[k + 1] = k + j;         // (k+1) + j - 1
        p2 = p1; p1 = j;
      }
    }
  }
  sb[T_ - 1] = T_ - 1;             // mindex[T] == 1
}

// Kernel C: one workgroup per (b, group of R_ channels). A single 2D TDM
// descriptor stages R_ contiguous 32KB rows into LDS; the jittered gather is
// served from LDS, reusing each srcidx int4 across all R_ rows; coalesced
// b128 stores out.
__global__ void __launch_bounds__(256)
jitter_gather_kernel(const float* __restrict__ x,
                     const int* __restrict__ srcidx,
                     float* __restrict__ out) {
  __shared__ float rows[R_][T_];         // 128 KB
  const int rowid0 = blockIdx.x * R_;    // first (b,i) row of this group
  const int b = rowid0 >> 8;             // I_ == 256, R_ divides 256
  const float* xtile = x + (size_t)rowid0 * T_;

  if (threadIdx.x < 32) {                // wave 0 issues the TDM DMA
    uint32_t lds_base = (uint32_t)(uintptr_t)(&rows[0][0]);
    uint64_t ga = (uint64_t)(uintptr_t)xtile;

    uint32x4_t_ g0;
    g0.x = 1u;                                   // count=1, user descriptor
    g0.y = lds_base;                             // lds_addr (bytes)
    g0.z = (uint32_t)ga;                         // global_addr[31:0]
    g0.w = (uint32_t)(ga >> 32) | (2u << 30);    // global_addr[56:32] | type=2

    int32x8_t_ g1;
    g1[0] = 0x00020000;                          // wg_mask=0, data_size=2 (4B)
    g1[1] = (int)((T_ & 0xFFFF) << 16);          // tensor_dim0[15:0]
    g1[2] = (int)((T_ >> 16) | ((unsigned)R_ << 16)); // tensor_dim0 hi | tensor_dim1=R_
    g1[3] = (int)(T_ << 16);                     // tile_dim0 = 8192
    g1[4] = R_;                                  // tile_dim1 = R_, tile_dim2 = 0
    g1[5] = T_;                                  // tensor_dim0_stride = 8192
    g1[6] = 0;
    g1[7] = 0;

    int32x4_t_ gz4 = {0, 0, 0, 0};
    int32x8_t_ gz8 = {0, 0, 0, 0, 0, 0, 0, 0};
    __builtin_amdgcn_tensor_load_to_lds(g0, g1, gz4, gz4, gz8, 0);
    __builtin_amdgcn_s_wait_tensorcnt(0);
  }
  __syncthreads();

  const int4* sidx = (const int4*)(srcidx + (size_t)b * T_);
  for (int q = threadIdx.x; q < T_ / 4; q += 256) {
    int4 s = sidx[q];
#pragma unroll
    for (int r = 0; r < R_; ++r) {
      float4 v;
      v.x = rows[r][s.x];
      v.y = rows[r][s.y];
      v.z = rows[r][s.z];
      v.w = rows[r][s.w];
      ((float4*)(out + (size_t)(rowid0 + r) * T_))[q] = v;
    }
  }
}

extern "C" void kernel_launch(void* const* d_in, const int* in_sizes, int n_in,
                              void* d_out, int out_size, void* d_ws, size_t ws_size,
                              hipStream_t stream) {
  const float* x     = (const float*)d_in[0];
  const float* probs = (const float*)d_in[1];
  float* out = (float*)d_out;

  uint8_t* cand  = (uint8_t*)d_ws;                      // 16*8192 = 128 KB
  int*     srcix = (int*)((char*)d_ws + 256 * 1024);    // 16*8192*4 = 512 KB

  int total = B_ * NDRAW;
  jitter_gumbel_kernel<<<(total + 255) / 256, 256, 0, stream>>>(probs, cand);
  jitter_scan_kernel<<<1, 32, 0, stream>>>(cand, srcix);
  jitter_gather_kernel<<<(B_ * I_) / R_, 256, 0, stream>>>(x, srcix, out);
}